// ENCConsLoss_12283606468280
// MI455X (gfx1250) — compile-verified
//
#include <hip/hip_runtime.h>

typedef __attribute__((ext_vector_type(2))) float v2f;
typedef __attribute__((ext_vector_type(8))) float v8f;

namespace {
constexpr int B_ = 8, T_ = 4, N_ = 784, C_ = 256, S_ = 16;
constexpr int M_   = 16;            // (B*T)/GF batched problems
constexpr int GN_  = 1568;          // GF*N rows per problem
constexpr int NT_  = GN_ / 16;      // 98 column tiles
constexpr int SSTR = GN_ + 2;       // 1570 (== 2 mod 8): strip row stride, bank-friendly
constexpr int CP_  = C_ + 2;        // 258: staged feature-tile row stride (pad banks)
constexpr float INV_TEMP   = 10.0f;             // 1/TEMP
constexpr float LOSS_SCALE = -(0.1f / 0.07f);   // -(TEMP/BASE_TEMP)
constexpr int ROWS_ = B_ * T_ * N_;             // 25088 feature rows per tensor
}

// ---------------------------------------------------------------------------
// Kernel 1: L2-normalize both feature tensors, relayout to (M, GN, C) f32.
// One wave per 256-wide row. Also zeroes the scalar output.
// ---------------------------------------------------------------------------
__global__ __launch_bounds__(256) void norm_kernel(
    const float* __restrict__ ft, const float* __restrict__ fc,
    float* __restrict__ ftn, float* __restrict__ fcn, float* __restrict__ out) {
  if (blockIdx.x == 0 && threadIdx.x == 0) out[0] = 0.f;
  const int wave = blockIdx.x * (blockDim.x >> 5) + (threadIdx.x >> 5);
  const int lane = threadIdx.x & 31;
  if (wave >= 2 * ROWS_) return;
  const int tensor = (wave >= ROWS_) ? 1 : 0;
  const int row = tensor ? (wave - ROWS_) : wave;
  const float* src = (tensor ? fc : ft) + (size_t)row * C_;

  const int b = row / (T_ * N_);
  const int rem = row % (T_ * N_);
  const int t = rem / N_;
  const int n = rem % N_;
  const int m = b * 2 + (t >> 1);
  const int gn = (t & 1) * N_ + n;
  float* dst = (tensor ? fcn : ftn) + ((size_t)m * GN_ + gn) * C_;

  float v[8];
  float ss = 0.f;
#pragma unroll
  for (int q = 0; q < 8; ++q) {
    v[q] = src[lane + 32 * q];
    ss += v[q] * v[q];
  }
#pragma unroll
  for (int d = 16; d >= 1; d >>= 1) ss += __shfl_xor(ss, d, 32);
  const float s = 1.0f / fmaxf(sqrtf(ss), 1e-12f);
#pragma unroll
  for (int q = 0; q < 8; ++q) dst[lane + 32 * q] = v[q] * s;
}

// ---------------------------------------------------------------------------
// Kernel 2: argmax over S of dec_masks, relayout to (M, GN) int.
// ---------------------------------------------------------------------------
__global__ __launch_bounds__(256) void argmax_kernel(
    const float* __restrict__ dm, int* __restrict__ amax) {
  const int idx = blockIdx.x * blockDim.x + threadIdx.x;
  if (idx >= B_ * T_ * N_) return;
  const int bt = idx / N_;
  const int n = idx % N_;
  const float* p = dm + ((size_t)bt * S_) * N_ + n;
  float best = p[0];
  int bi = 0;
#pragma unroll
  for (int s = 1; s < S_; ++s) {
    const float v = p[(size_t)s * N_];
    if (v > best) { best = v; bi = s; }  // first-occurrence semantics (strict >)
  }
  amax[(bt >> 1) * GN_ + (bt & 1) * N_ + n] = bi;
}

// ---------------------------------------------------------------------------
// Kernel 3: fused strip kernel. Block = (i-tile, m). 256 threads = 8 waves.
//   Phase A: WMMA f32 selfsim strip (16 x 1568) -> LDS (resident, ~100 KB)
//   Phase T: per-row 16th-largest threshold (wave max-extraction)
//   Phase B: WMMA f32 logits pass, fused masked accumulation (5 sums/row)
//   Phase F: per-row loss, atomicAdd into scalar output
// ---------------------------------------------------------------------------
__global__ __launch_bounds__(256, 1) void loss_kernel(
    const float* __restrict__ ftn, const float* __restrict__ fcn,
    const int* __restrict__ amax, float* __restrict__ out) {
  extern __shared__ float smem[];
  float* sstrip  = smem;                    // 16*SSTR floats (selfsim strip)
  float* sfci    = sstrip + 16 * SSTR;      // 16*CP_ (criterion i-tile)
  float* sfti    = sfci + 16 * CP_;         // 16*CP_ (trainable i-tile)
  float* sthresh = sfti + 16 * CP_;         // 16
  float* srow    = sthresh + 16;            // 16*5 row accumulators
  int*   samax   = (int*)(srow + 80);       // GN_ ints

  const int m  = blockIdx.y;
  const int i0 = blockIdx.x * 16;
  const int tid  = threadIdx.x;
  const int lane = tid & 31;
  const int wid  = tid >> 5;
  const int r = lane & 15;   // row-within-tile operand lane / output column
  const int h = lane >> 4;   // lane half

  const float* fcm = fcn + (size_t)m * GN_ * C_;
  const float* ftm = ftn + (size_t)m * GN_ * C_;

  // ---- Phase 0: stage i-tiles (padded stride) + argmax row + zero accums ----
  for (int x = tid; x < 16 * C_; x += 256) {
    const int rw = x >> 8, cl = x & 255;
    sfci[rw * CP_ + cl] = fcm[(size_t)(i0 + rw) * C_ + cl];
    sfti[rw * CP_ + cl] = ftm[(size_t)(i0 + rw) * C_ + cl];
  }
  for (int x = tid; x < GN_; x += 256) samax[x] = amax[m * GN_ + x];
  if (tid < 80) srow[tid] = 0.f;
  __syncthreads();

  // ---- Phase A: selfsim strip via V_WMMA_F32_16X16X4_F32 ----
  for (int jt = wid; jt < NT_; jt += 8) {
    const float* bptr = fcm + (size_t)(jt * 16 + r) * C_ + 2 * h;
    const float* aptr = sfci + r * CP_ + 2 * h;
    __builtin_prefetch(bptr + (size_t)8 * 16 * C_, 0, 0);  // next tile for this wave
    v8f acc = {0.f, 0.f, 0.f, 0.f, 0.f, 0.f, 0.f, 0.f};
#pragma unroll 8
    for (int k = 0; k < C_; k += 4) {
      const v2f a = *(const v2f*)(aptr + k);
      const v2f b = *(const v2f*)(bptr + k);
      acc = __builtin_amdgcn_wmma_f32_16x16x4_f32(false, a, false, b, (short)0,
                                                  acc, false, false);
    }
#pragma unroll
    for (int v = 0; v < 8; ++v)
      sstrip[(v + 8 * h) * SSTR + jt * 16 + r] = acc[v];
  }
  __syncthreads();

  // ---- Phase T: 16th-largest per row (16 max-extractions over LDS row) ----
  for (int rr = wid; rr < 16; rr += 8) {
    const float* rowp = sstrip + rr * SSTR;
    float bound = 3.4e38f;
    for (int iter = 0; iter < 16; ++iter) {
      float mx = -3.4e38f;
      for (int j = lane; j < GN_; j += 32) {
        const float v = rowp[j];
        mx = (v < bound && v > mx) ? v : mx;
      }
#pragma unroll
      for (int d = 16; d >= 1; d >>= 1) mx = fmaxf(mx, __shfl_xor(mx, d, 32));
      bound = mx;
    }
    if (lane == 0) sthresh[rr] = bound;
  }
  __syncthreads();

  // ---- Phase B: logits pass, fused masked accumulation ----
  float aExp[8], aPs[8], aPc[8], aSs[8], aSc[8];
  float myTh[8];
  int myAm[8];
#pragma unroll
  for (int v = 0; v < 8; ++v) {
    myTh[v] = sthresh[v + 8 * h];
    myAm[v] = samax[i0 + v + 8 * h];
    aExp[v] = aPs[v] = aPc[v] = aSs[v] = aSc[v] = 0.f;
  }

  for (int jt = wid; jt < NT_; jt += 8) {
    const float* bptr = ftm + (size_t)(jt * 16 + r) * C_ + 2 * h;
    const float* aptr = sfti + r * CP_ + 2 * h;
    __builtin_prefetch(bptr + (size_t)8 * 16 * C_, 0, 0);
    v8f acc = {0.f, 0.f, 0.f, 0.f, 0.f, 0.f, 0.f, 0.f};
#pragma unroll 8
    for (int k = 0; k < C_; k += 4) {
      const v2f a = *(const v2f*)(aptr + k);
      const v2f b = *(const v2f*)(bptr + k);
      acc = __builtin_amdgcn_wmma_f32_16x16x4_f32(false, a, false, b, (short)0,
                                                  acc, false, false);
    }
    const int j = jt * 16 + r;
    const int aj = samax[j];
#pragma unroll
    for (int v = 0; v < 8; ++v) {
      const int irow = i0 + v + 8 * h;
      const float logit = acc[v] * INV_TEMP;
      const bool diag = (irow == j);
      aExp[v] += diag ? 0.f : __expf(logit);
      const float ssv = sstrip[(v + 8 * h) * SSTR + j];
      const float p  = (!diag && ssv >= myTh[v]) ? 1.f : 0.f;
      const float sm = (!diag && aj == myAm[v]) ? 1.f : 0.f;
      aPs[v] += p * logit;
      aPc[v] += p;
      aSs[v] += sm * logit;
      aSc[v] += sm;
    }
  }

  // reduce across the 16 lanes sharing each row-half, then LDS atomics
#pragma unroll
  for (int v = 0; v < 8; ++v) {
#pragma unroll
    for (int d = 8; d >= 1; d >>= 1) {
      aExp[v] += __shfl_xor(aExp[v], d, 32);
      aPs[v]  += __shfl_xor(aPs[v], d, 32);
      aPc[v]  += __shfl_xor(aPc[v], d, 32);
      aSs[v]  += __shfl_xor(aSs[v], d, 32);
      aSc[v]  += __shfl_xor(aSc[v], d, 32);
    }
  }
  if (r == 0) {
#pragma unroll
    for (int v = 0; v < 8; ++v) {
      const int rr = v + 8 * h;
      atomicAdd(&srow[rr * 5 + 0], aExp[v]);
      atomicAdd(&srow[rr * 5 + 1], aPs[v]);
      atomicAdd(&srow[rr * 5 + 2], aPc[v]);
      atomicAdd(&srow[rr * 5 + 3], aSs[v]);
      atomicAdd(&srow[rr * 5 + 4], aSc[v]);
    }
  }
  __syncthreads();

  // ---- Phase F: per-row loss contribution ----
  if (tid < 16) {
    const float se  = srow[tid * 5 + 0];
    const float lse = logf(se);
    const float ps = srow[tid * 5 + 1], pc = srow[tid * 5 + 2];
    const float ms = srow[tid * 5 + 3], mc = srow[tid * 5 + 4];
    const float posL = LOSS_SCALE * (ps - pc * lse) / (pc + 1e-8f);
    const float semL = LOSS_SCALE * (ms - mc * lse) / (mc + 1e-8f);
    atomicAdd(out, (posL + semL) * (1.0f / (float)(M_ * GN_ * 2)));
  }
}

// ---------------------------------------------------------------------------
extern "C" void kernel_launch(void* const* d_in, const int* in_sizes, int n_in,
                              void* d_out, int out_size, void* d_ws,
                              size_t ws_size, hipStream_t stream) {
  (void)in_sizes; (void)n_in; (void)out_size; (void)ws_size;
  const float* ft = (const float*)d_in[0];  // feat_trainable (B,T,N,C) f32
  const float* fc = (const float*)d_in[1];  // feat_criterion (B,T,N,D) f32
  const float* dm = (const float*)d_in[2];  // dec_masks (B,T,S,N) f32
  float* out = (float*)d_out;

  float* ftn = (float*)d_ws;                            // (16,1568,256) f32
  float* fcn = ftn + (size_t)M_ * GN_ * C_;             // (16,1568,256) f32
  int* amax = (int*)(fcn + (size_t)M_ * GN_ * C_);      // (16,1568) int

  // 2*25088 waves, 8 waves/block
  norm_kernel<<<(2 * ROWS_) / 8, 256, 0, stream>>>(ft, fc, ftn, fcn, out);
  argmax_kernel<<<(B_ * T_ * N_ + 255) / 256, 256, 0, stream>>>(dm, amax);

  const size_t shbytes =
      (size_t)(16 * SSTR + 2 * 16 * CP_ + 16 + 80 + GN_) * sizeof(float);
  dim3 grid(NT_, M_);
  loss_kernel<<<grid, 256, shbytes, stream>>>(ftn, fcn, amax, out);
}